// attention_30193620091021
// MI455X (gfx1250) — compile-verified
//
#include <hip/hip_runtime.h>

// Problem constants (match reference)
#define B_ 8
#define T_ 4096
#define E_ 64
#define D_ 8

typedef __attribute__((ext_vector_type(2))) float v2f;
typedef __attribute__((ext_vector_type(4))) float v4f;
typedef __attribute__((ext_vector_type(8))) float v8f;

// ---------------------------------------------------------------------------
// Kernel 1: fused q/k/v projection.  One thread per (b,t) row.
// ---------------------------------------------------------------------------
__global__ __launch_bounds__(256) void qkv_proj_kernel(
    const float* __restrict__ x,
    const float* __restrict__ Wq, const float* __restrict__ bq,
    const float* __restrict__ Wk, const float* __restrict__ bk,
    const float* __restrict__ Wv, const float* __restrict__ bv,
    float* __restrict__ q, float* __restrict__ k, float* __restrict__ v)
{
    int row = blockIdx.x * blockDim.x + threadIdx.x;
    if (row >= B_ * T_) return;
    const float* xr = x + (size_t)row * E_;

    float aq[D_], ak[D_], av[D_];
    #pragma unroll
    for (int d = 0; d < D_; ++d) { aq[d] = bq[d]; ak[d] = bk[d]; av[d] = bv[d]; }

    #pragma unroll 4
    for (int e = 0; e < E_; ++e) {
        float xe = xr[e];
        #pragma unroll
        for (int d = 0; d < D_; ++d) {
            aq[d] = fmaf(xe, Wq[e * D_ + d], aq[d]);
            ak[d] = fmaf(xe, Wk[e * D_ + d], ak[d]);
            av[d] = fmaf(xe, Wv[e * D_ + d], av[d]);
        }
    }

    float* qr = q + (size_t)row * D_;
    float* kr = k + (size_t)row * D_;
    float* vr = v + (size_t)row * D_;
    *(v4f*)(qr)     = (v4f){aq[0], aq[1], aq[2], aq[3]};
    *(v4f*)(qr + 4) = (v4f){aq[4], aq[5], aq[6], aq[7]};
    *(v4f*)(kr)     = (v4f){ak[0], ak[1], ak[2], ak[3]};
    *(v4f*)(kr + 4) = (v4f){ak[4], ak[5], ak[6], ak[7]};
    *(v4f*)(vr)     = (v4f){av[0], av[1], av[2], av[3]};
    *(v4f*)(vr + 4) = (v4f){av[4], av[5], av[6], av[7]};
}

// ---------------------------------------------------------------------------
// One 16(j) x 16(i) tile of the flash loop, transposed score layout:
//   lanes 0..15 : reg r = tile[j=r,   i=lane]
//   lanes 16..31: reg r = tile[j=r+8, i=lane-16]
// q operand is pre-scaled by (1/sqrt(8))*log2(e), so all exponentials are
// raw v_exp_f32 (exp2).  DIAG adds the causal j<=i mask (only needed on the
// diagonal tile).  The w==0 -> -inf reference quirk is the (s != 0) test
// (positive scaling preserves exact zeros).
// ---------------------------------------------------------------------------
template<bool DIAG>
__device__ __forceinline__ void attn_tile(
    int jg0, int i, bool hi, int l15, bool prefetch,
    const float* __restrict__ kb, const float* __restrict__ vb,
    v2f bq0, v2f bq1, float& m, float& l, v8f& O)
{
    const int j  = jg0 + l15;
    const int db = hi ? 2 : 0;

    // ---- scores: D[m=j, n=i] = sum_d K2[d,j] * (q[i,d]*SC) ----
    v2f a0, a1;
    a0.x = kb[(db + 0) * T_ + j]; a0.y = kb[(db + 1) * T_ + j];
    a1.x = kb[(db + 4) * T_ + j]; a1.y = kb[(db + 5) * T_ + j];
    if (prefetch) {
        __builtin_prefetch(&kb[db * T_ + j + 16], 0, 3);        // global_prefetch_b8
        __builtin_prefetch(&vb[(jg0 + 16) * D_ + l15], 0, 3);
    }

    v8f s = {};
    s = __builtin_amdgcn_wmma_f32_16x16x4_f32(false, a0, false, bq0,
                                              (short)0, s, false, false);
    s = __builtin_amdgcn_wmma_f32_16x16x4_f32(false, a1, false, bq1,
                                              (short)0, s, false, false);

    // ---- mask + online softmax (log2 domain) ----
    float p[8];
    float mt = -__builtin_inff();
    #pragma unroll
    for (int r = 0; r < 8; ++r) {
        float sv = s[r];
        bool valid = (sv != 0.0f);
        if (DIAG) {
            int js = jg0 + r + (hi ? 8 : 0);
            valid = valid && (js <= i);
        }
        float lg = valid ? sv : -__builtin_inff();
        p[r] = lg;
        mt = fmaxf(mt, lg);
    }
    mt = fmaxf(mt, __shfl_xor(mt, 16, 32));          // combine half-wave partners
    float mnew  = fmaxf(m, mt);
    float alpha = __builtin_amdgcn_exp2f(m - mnew);
    m = mnew;

    float ls = 0.0f;
    #pragma unroll
    for (int r = 0; r < 8; ++r) {
        p[r] = __builtin_amdgcn_exp2f(p[r] - mnew);
        ls += p[r];
    }
    ls += __shfl_xor(ls, 16, 32);
    l = l * alpha + ls;
    #pragma unroll
    for (int r = 0; r < 8; ++r) O[r] *= alpha;

    // ---- PV: O[d,i] += sum_j V[j,d] * P[j,i]  (4 chained 16x16x4 WMMAs) ----
    // Rebuild B (P chunks) from probability regs: 8 xor-16 shuffles.
    float t0 = __shfl_xor(p[0], 16, 32), t1 = __shfl_xor(p[1], 16, 32);
    float t2 = __shfl_xor(p[2], 16, 32), t3 = __shfl_xor(p[3], 16, 32);
    float t4 = __shfl_xor(p[4], 16, 32), t5 = __shfl_xor(p[5], 16, 32);
    float t6 = __shfl_xor(p[6], 16, 32), t7 = __shfl_xor(p[7], 16, 32);

    // A rows d=8..15 read duplicated (finite) V values; the resulting D rows
    // 8..15 are never stored and cannot contaminate rows 0..7.
    const int dc = l15 & (D_ - 1);

    #pragma unroll
    for (int c = 0; c < 4; ++c) {
        int jb = jg0 + 4 * c + (hi ? 2 : 0);
        v2f av2;
        av2.x = vb[(jb + 0) * D_ + dc];
        av2.y = vb[(jb + 1) * D_ + dc];

        v2f bp;
        if      (c == 0) { bp.x = hi ? t2 : p[0]; bp.y = hi ? t3 : p[1]; }
        else if (c == 1) { bp.x = hi ? t6 : p[4]; bp.y = hi ? t7 : p[5]; }
        else if (c == 2) { bp.x = hi ? p[2] : t0; bp.y = hi ? p[3] : t1; }
        else             { bp.x = hi ? p[6] : t4; bp.y = hi ? p[7] : t5; }

        O = __builtin_amdgcn_wmma_f32_16x16x4_f32(false, av2, false, bp,
                                                  (short)0, O, false, false);
    }
}

// ---------------------------------------------------------------------------
// Kernel 2: flash-style causal attention with the reference's reshape quirk
// (K2[d,j] = kflat[d*T + j]).  One wave per 16 query rows, 8 waves per block.
// ---------------------------------------------------------------------------
__global__ __launch_bounds__(256) void causal_attn_kernel(
    const float* __restrict__ qf, const float* __restrict__ kf,
    const float* __restrict__ vf, float* __restrict__ out)
{
    const int lane = threadIdx.x & 31;
    const int wave = threadIdx.x >> 5;
    const int b    = blockIdx.y;
    const int i0   = blockIdx.x * 128 + wave * 16;   // 16 query rows per wave

    const float* qb = qf + (size_t)b * T_ * D_;
    const float* kb = kf + (size_t)b * T_ * D_;      // viewed as K2[D_][T_]
    const float* vb = vf + (size_t)b * T_ * D_;

    const int  l15 = lane & 15;
    const bool hi  = lane >= 16;
    const int  i   = i0 + l15;                       // query row for this lane

    // logits computed directly in log2 domain: SC = (1/sqrt(8)) * log2(e)
    const float SC = 0.35355339059327373f * 1.4426950408889634f;

    // Loop-invariant, pre-scaled B operand for scores: B[k=d, n=i] = q[i,d]*SC
    const int kb0 = hi ? 2 : 0;
    v2f bq0, bq1;
    bq0.x = qb[i * D_ + kb0 + 0] * SC; bq0.y = qb[i * D_ + kb0 + 1] * SC;
    bq1.x = qb[i * D_ + kb0 + 4] * SC; bq1.y = qb[i * D_ + kb0 + 5] * SC;

    float m = -__builtin_inff();
    float l = 0.0f;
    v8f O = {};                                      // O[d=r (+8 for hi), i]

    // Main loop: tiles strictly below the diagonal (no causal compare).
    const int nmain = i0 >> 4;
    for (int jt = 0; jt < nmain; ++jt)
        attn_tile<false>(jt << 4, i, hi, l15, true, kb, vb, bq0, bq1, m, l, O);

    // Diagonal tile: needs the j<=i mask.
    attn_tile<true>(i0, i, hi, l15, false, kb, vb, bq0, bq1, m, l, O);

    // ---- epilogue: lane L<16 owns output row i0+L, 8 contiguous floats ----
    if (!hi) {
        float inv = 1.0f / l;
        float* orow = out + ((size_t)b * T_ + i) * D_;
        *(v4f*)(orow)     = (v4f){O[0] * inv, O[1] * inv, O[2] * inv, O[3] * inv};
        *(v4f*)(orow + 4) = (v4f){O[4] * inv, O[5] * inv, O[6] * inv, O[7] * inv};
    }
}

// ---------------------------------------------------------------------------
extern "C" void kernel_launch(void* const* d_in, const int* in_sizes, int n_in,
                              void* d_out, int out_size, void* d_ws, size_t ws_size,
                              hipStream_t stream)
{
    const float* x  = (const float*)d_in[0];
    const float* Wq = (const float*)d_in[1];
    const float* bq = (const float*)d_in[2];
    const float* Wk = (const float*)d_in[3];
    const float* bk = (const float*)d_in[4];
    const float* Wv = (const float*)d_in[5];
    const float* bv = (const float*)d_in[6];
    float* out = (float*)d_out;

    float* q = (float*)d_ws;               // 3 * B*T*D floats = 3 MB scratch
    float* k = q + (size_t)B_ * T_ * D_;
    float* v = k + (size_t)B_ * T_ * D_;

    qkv_proj_kernel<<<(B_ * T_ + 255) / 256, 256, 0, stream>>>(
        x, Wq, bq, Wk, bk, Wv, bv, q, k, v);

    causal_attn_kernel<<<dim3(T_ / 128, B_), 256, 0, stream>>>(q, k, v, out);
}